// IndRNN_26319559589930
// MI455X (gfx1250) — compile-verified
//
#include <hip/hip_runtime.h>
#include <hip/hip_bf16.h>

// IndRNN, 2 layers fused: h1 = relu(x + w1*h1); h2 = relu(h1 + w2*h2); out = h2.
// Elementwise recurrence over T per (batch,hidden) channel; relu => no
// associative scan, time must be walked serially per channel.
// Memory-bound: 128MB in + 128MB out => ~11us roofline at 23.3 TB/s.
// Strategy: one thread per channel, both layer states in registers (halves
// traffic vs. two passes). x streamed HBM -> LDS with CDNA5 async copies
// (global_load_async_to_lds_b128 / ASYNCcnt), depth-6 pipeline across 8 LDS
// tile buffers (96KB in flight per block) so the dependent fma/max chain
// never observes HBM latency. Touch-once streams use NT cache hints.

#define BLOCK 128                 // threads per block = channels per block (4 waves)
#define TT    32                  // timesteps per tile
#define NBUF  8                   // LDS tile buffers (128 KB of 320 KB/WGP)
#define DEPTH 6                   // tiles prefetched ahead (56 max ASYNCcnt <= 63)
#define IPT   (TT / 4)            // async-load instructions per tile per wave (=8)

#define WAIT_ASYNC(n) asm volatile("s_wait_asynccnt " #n ::: "memory")

template <int C_CT, int H_CT>
__global__ __launch_bounds__(BLOCK) void indrnn_fused_kernel(
    const float* __restrict__ x,     // [T, C]  (C = B*H, contiguous channels)
    const float* __restrict__ w_hh,  // [2, H]
    const float* __restrict__ h0,    // [2, C]
    float* __restrict__ out,         // [T, C]
    int T, int C_rt, int H_rt)
{
    const int C = C_CT ? C_CT : C_rt;
    const int H = H_CT ? H_CT : H_rt;

    __shared__ float smem[NBUF][TT * BLOCK];   // 128 KB total

    const int tid = threadIdx.x;
    const int c0  = blockIdx.x * BLOCK;        // first channel of this block
    const int c   = c0 + tid;                  // this thread's channel
    const int hh  = c % H;                     // folds to AND when H_CT is pow2

    const float w1 = w_hh[hh];
    const float w2 = w_hh[H + hh];
    float h1 = h0[c];
    float h2 = h0[C + c];

    // Load mapping: each lane moves 16B (4 channels) of one row per issue.
    // 32 lanes cover one 512B row; 4 waves x 1 row sub-index => 4 rows/issue.
    const int rBase = tid >> 5;                // wave-uniform row sub-index 0..3
    const int cin   = (tid & 31) * 4;          // float offset within row

    const int ntiles = (T + TT - 1) / TT;

    auto issue_tile = [&](int k) {
        const int buf = k & (NBUF - 1);
        const int t0  = k * TT;
        const uint32_t ldsBase =
            (uint32_t)(size_t)(const void*)&smem[buf][0];  // addr[31:0] == LDS offset
#pragma unroll
        for (int i = 0; i < IPT; ++i) {
            const int r = i * 4 + rBase;
            int t = t0 + r;
            t = (t < T) ? t : (T - 1);         // clamp keeps ASYNCcnt uniform per wave
            const float* gp = x + (size_t)t * C + (c0 + cin);
            const uint32_t lo = ldsBase + (uint32_t)((r * BLOCK + cin) * 4);
            asm volatile("global_load_async_to_lds_b128 %0, %1, off th:TH_LOAD_NT"
                         :: "v"(lo), "v"(gp)
                         : "memory");
        }
    };

    // Prologue: fill the pipeline DEPTH tiles deep.
#pragma unroll
    for (int k = 0; k < DEPTH; ++k)
        if (k < ntiles) issue_tile(k);

    for (int k = 0; k < ntiles; ++k) {
        const int kp = k + DEPTH;
        if (kp < ntiles) issue_tile(kp);

        // Drain own wave's ASYNCcnt down to the tiles still in flight beyond k.
        // Async loads complete in order; each tile is IPT(=8) ops per wave.
        const int rem = ntiles - 1 - k;
        const int p   = rem < DEPTH ? rem : DEPTH;
        switch (p) {
            case 6:  WAIT_ASYNC(48); break;
            case 5:  WAIT_ASYNC(40); break;
            case 4:  WAIT_ASYNC(32); break;
            case 3:  WAIT_ASYNC(24); break;
            case 2:  WAIT_ASYNC(16); break;
            case 1:  WAIT_ASYNC(8);  break;
            default: WAIT_ASYNC(0);  break;
        }
        __syncthreads();                       // tile k visible to all waves

        const float* sp = &smem[k & (NBUF - 1)][tid];
        const int t0 = k * TT;
        float* op = out + (size_t)t0 * C + c;

        if (t0 + TT <= T) {
#pragma unroll
            for (int r = 0; r < TT; ++r) {
                const float xv = sp[r * BLOCK];
                h1 = fmaxf(fmaf(h1, w1, xv), 0.0f);
                h2 = fmaxf(fmaf(h2, w2, h1), 0.0f);
                __builtin_nontemporal_store(h2, op + (size_t)r * C);
            }
        } else {
            for (int r = 0; r < T - t0; ++r) {
                const float xv = sp[r * BLOCK];
                h1 = fmaxf(fmaf(h1, w1, xv), 0.0f);
                h2 = fmaxf(fmaf(h2, w2, h1), 0.0f);
                __builtin_nontemporal_store(h2, op + (size_t)r * C);
            }
        }
        __syncthreads();                       // buffer reuse guard
    }
}

extern "C" void kernel_launch(void* const* d_in, const int* in_sizes, int n_in,
                              void* d_out, int out_size, void* d_ws, size_t ws_size,
                              hipStream_t stream) {
    const float* x    = (const float*)d_in[0];   // [T, B, H]
    const float* w_hh = (const float*)d_in[1];   // [2, H]
    const float* h0   = (const float*)d_in[2];   // [2, B, H]
    float*       out  = (float*)d_out;           // [T, B, H]

    const int H = in_sizes[1] / 2;               // N_LAYER = 2
    const int C = in_sizes[2] / 2;               // B*H
    const int T = in_sizes[0] / C;

    const int nblocks = C / BLOCK;               // C = 16384 -> 128 blocks
    if (C == 16384 && H == 512) {
        indrnn_fused_kernel<16384, 512><<<nblocks, BLOCK, 0, stream>>>(
            x, w_hh, h0, out, T, C, H);
    } else {
        indrnn_fused_kernel<0, 0><<<nblocks, BLOCK, 0, stream>>>(
            x, w_hh, h0, out, T, C, H);
    }
}